// S4RecurrentEfficient_16123307229762
// MI455X (gfx1250) — compile-verified
//
#include <hip/hip_runtime.h>
#include <hip/hip_bf16.h>

#define NSTATE 64
#define SEQL   4096
#define NBATCH 16
#define DMODEL 512

typedef float v2f __attribute__((ext_vector_type(2)));
typedef float v4f __attribute__((ext_vector_type(4)));
typedef float v8f __attribute__((ext_vector_type(8)));

// ---------------------------------------------------------------------------
// Kernel 1 (1 block, 256 threads):
//   dt      = clip(exp(log_dt), 1e-3, 1e-1)
//   A_bar   = expm(dt*A)            (Taylor, 24 terms, converged for ||dA||<0.4)
//   Bb      = dt * phi1(dt*A) @ B   (phi1 = sum dA^k/(k+1)!   -- no inverse!)
//   P16     = A_bar^16, P256 = A_bar^256 (repeated squaring)
//   seeds[g]= C @ A_bar^(256*g), g=0..15
// ---------------------------------------------------------------------------
__global__ __launch_bounds__(256) void s4_setup_kernel(
    const float* __restrict__ A, const float* __restrict__ Bv,
    const float* __restrict__ Cv, const float* __restrict__ log_dt,
    float* __restrict__ wsAbar, float* __restrict__ wsP16,
    float* __restrict__ wsBb,   float* __restrict__ wsSeeds)
{
  __shared__ float dAs[NSTATE * NSTATE];
  __shared__ float T1 [NSTATE * NSTATE];
  __shared__ float T2 [NSTATE * NSTATE];
  __shared__ float Ab [NSTATE * NSTATE];
  __shared__ float Bacc[NSTATE];
  __shared__ float wrow[2][NSTATE];
  __shared__ float sdt;

  const int tid = threadIdx.x;
  if (tid == 0) {
    float dt = expf(log_dt[0]);
    sdt = fminf(fmaxf(dt, 0.001f), 0.1f);
  }
  __syncthreads();
  const float dt = sdt;

  for (int e = tid; e < NSTATE * NSTATE; e += 256) dAs[e] = dt * A[e];
  __syncthreads();

  // j = 1 term: M1 = dA.  Ab = I + dA.  Bacc = B + (dA @ B)/2
  for (int e = tid; e < NSTATE * NSTATE; e += 256) {
    int r = e >> 6, c = e & 63;
    Ab[e] = ((r == c) ? 1.0f : 0.0f) + dAs[e];
    T1[e] = dAs[e];
  }
  if (tid < NSTATE) {
    float s = 0.0f;
    for (int q = 0; q < NSTATE; ++q) s += dAs[tid * NSTATE + q] * Bv[q];
    Bacc[tid] = Bv[tid] + 0.5f * s;
  }
  __syncthreads();

  float* M  = T1;
  float* Mn = T2;
  for (int j = 2; j <= 24; ++j) {
    const float inv_j = 1.0f / (float)j;
    for (int e = tid; e < NSTATE * NSTATE; e += 256) {
      int r = e >> 6, c = e & 63;
      float s = 0.0f;
      const float* mr = &M[r * NSTATE];
      for (int q = 0; q < NSTATE; ++q) s += mr[q] * dAs[q * NSTATE + c];
      Mn[e] = s * inv_j;          // M_j = M_{j-1} @ dA / j  (= dA^j / j!)
    }
    __syncthreads();
    const float inv_j1 = 1.0f / (float)(j + 1);
    for (int e = tid; e < NSTATE * NSTATE; e += 256) Ab[e] += Mn[e];
    if (tid < NSTATE) {
      float s = 0.0f;
      const float* mr = &Mn[tid * NSTATE];
      for (int q = 0; q < NSTATE; ++q) s += mr[q] * Bv[q];
      Bacc[tid] += s * inv_j1;    // phi1 contribution: M_j @ B / (j+1)
    }
    __syncthreads();
    float* t = M; M = Mn; Mn = t;
  }

  if (tid < NSTATE) wsBb[tid] = dt * Bacc[tid];
  for (int e = tid; e < NSTATE * NSTATE; e += 256) { wsAbar[e] = Ab[e]; T1[e] = Ab[e]; }
  __syncthreads();

  // Repeated squaring: after 4 -> A^16, after 8 -> A^256
  float* P  = T1;
  float* Pn = T2;
  for (int sq = 0; sq < 8; ++sq) {
    for (int e = tid; e < NSTATE * NSTATE; e += 256) {
      int r = e >> 6, c = e & 63;
      float s = 0.0f;
      const float* pr = &P[r * NSTATE];
      for (int q = 0; q < NSTATE; ++q) s += pr[q] * P[q * NSTATE + c];
      Pn[e] = s;
    }
    __syncthreads();
    { float* t = P; P = Pn; Pn = t; }
    if (sq == 3) {
      for (int e = tid; e < NSTATE * NSTATE; e += 256) wsP16[e] = P[e];
      __syncthreads();
    }
  }
  // P == A_bar^256 now. Seed rows: seeds[g] = C @ P^g
  if (tid < NSTATE) wrow[0][tid] = Cv[tid];
  __syncthreads();
  int cur = 0;
  for (int g = 0; g < 16; ++g) {
    if (tid < NSTATE) wsSeeds[g * NSTATE + tid] = wrow[cur][tid];
    if (g < 15) {
      if (tid < NSTATE) {
        float s = 0.0f;
        for (int q = 0; q < NSTATE; ++q) s += wrow[cur][q] * P[q * NSTATE + tid];
        wrow[1 - cur][tid] = s;
      }
      __syncthreads();
      cur = 1 - cur;
    }
  }
}

// ---------------------------------------------------------------------------
// Kernel 2 (16 blocks, 256 threads): block g produces k[256g .. 256g+255],
// k[t] = (C A_bar^t) . Bb.  Builds 16 consecutive rows from the seed, then
// advances 16 t-steps at a time with W <- W @ P16.
// ---------------------------------------------------------------------------
__global__ __launch_bounds__(256) void s4_kgen_kernel(
    const float* __restrict__ wsAbar, const float* __restrict__ wsP16,
    const float* __restrict__ wsBb,   const float* __restrict__ wsSeeds,
    float* __restrict__ wsK)
{
  __shared__ float Ab [NSTATE * NSTATE];
  __shared__ float P16[NSTATE * NSTATE];
  __shared__ float Bb [NSTATE];
  __shared__ float W  [2][16 * NSTATE];

  const int tid = threadIdx.x;
  const int g   = blockIdx.x;

  for (int e = tid; e < NSTATE * NSTATE; e += 256) { Ab[e] = wsAbar[e]; P16[e] = wsP16[e]; }
  if (tid < NSTATE) { Bb[tid] = wsBb[tid]; W[0][tid] = wsSeeds[g * NSTATE + tid]; }
  __syncthreads();

  for (int j = 1; j < 16; ++j) {          // rows t = 256g + j
    if (tid < NSTATE) {
      float s = 0.0f;
      const float* wp = &W[0][(j - 1) * NSTATE];
      for (int q = 0; q < NSTATE; ++q) s += wp[q] * Ab[q * NSTATE + tid];
      W[0][j * NSTATE + tid] = s;
    }
    __syncthreads();
  }

  int cur = 0;
  for (int c = 0; c < 16; ++c) {
    if (tid < 16) {
      float s = 0.0f;
      const float* wp = &W[cur][tid * NSTATE];
      for (int q = 0; q < NSTATE; ++q) s += wp[q] * Bb[q];
      wsK[g * 256 + c * 16 + tid] = s;
    }
    if (c < 15) {                          // W <- W @ P16 (advance 16 steps)
      for (int e = tid; e < 16 * NSTATE; e += 256) {
        int r = e >> 6, col = e & 63;
        float s = 0.0f;
        const float* wp = &W[cur][r * NSTATE];
        for (int q = 0; q < NSTATE; ++q) s += wp[q] * P16[q * NSTATE + col];
        W[1 - cur][e] = s;
      }
      __syncthreads();
      cur = 1 - cur;
    }
  }
}

// ---------------------------------------------------------------------------
// Kernel 3: u_proj[row] = u[row,:] . W_in.  One wave32 per row; four b128
// non-temporal loads per lane (u is a 134 MB one-shot stream -- keep it out
// of L2 so the reused up/k/yp working set stays resident), shuffle reduce.
// ---------------------------------------------------------------------------
__global__ __launch_bounds__(256) void s4_uproj_kernel(
    const float* __restrict__ u, const float* __restrict__ Win,
    float* __restrict__ up)
{
  const int gid  = blockIdx.x * 256 + threadIdx.x;
  const int row  = gid >> 5;
  const int lane = gid & 31;
  if (row >= NBATCH * SEQL) return;
  const v4f* ur4 = (const v4f*)(u + (size_t)row * DMODEL);
  const v4f* wi4 = (const v4f*)Win;
  float s = 0.0f;
#pragma unroll
  for (int i = 0; i < DMODEL / 128; ++i) {
    int idx = lane + 32 * i;               // 512B contiguous per 32-lane chunk
    v4f a = __builtin_nontemporal_load(ur4 + idx);
    v4f w = wi4[idx];
    s += a.x * w.x + a.y * w.y + a.z * w.z + a.w * w.w;
  }
#pragma unroll
  for (int off = 16; off > 0; off >>= 1) s += __shfl_xor(s, off, 32);
  if (lane == 0) up[row] = s;
}

// ---------------------------------------------------------------------------
// Kernel 4: causal convolution via V_WMMA_F32_16X16X4_F32, split-K over 4
// waves per block.  One block per 16-wide time tile.  Wave w covers
// s = 16*it + 4w + [0..3] -- an exact partition of [0, smax).
// The causal mask (k-index >= 0) can only trigger in the final 16-wide
// s-round: for round `it` the worst lane (t=t0, w=3, kb=2, +1) needs
// t0-2-sk >= 0, true for all it < t0/16.  So run t0/16 mask-free rounds
// (b64 A-load + 2 ds_loads + WMMA only) and one masked boundary round.
// Per-wave loop bounds are uniform => EXEC all-ones around WMMA.
// Fuses yprime = Y + D*up after a lane-wise LDS reduction.
// ---------------------------------------------------------------------------
__global__ __launch_bounds__(128) void s4_conv_kernel(
    const float* __restrict__ up, const float* __restrict__ wsK,
    const float* __restrict__ Dptr, float* __restrict__ yp)
{
  __shared__ float ks[SEQL];
  __shared__ float part[4][8][32];

  const int tid  = threadIdx.x;
  const int lane = tid & 31;
  const int w    = tid >> 5;
  const int t0   = blockIdx.x * 16;

  {
    const float4* src = (const float4*)wsK;
    float4*       dst = (float4*)ks;
    for (int i = tid; i < SEQL / 4; i += 128) dst[i] = src[i];
  }
  __syncthreads();

  const int m  = lane & 15;                // A rows: batch (M dim)
  const int kb = (lane >> 4) * 2;          // K sub-position per half-wave
  const int t  = t0 + (lane & 15);         // B cols: time (N dim)
  const float* uprow = up + m * SEQL;

  v8f acc = {0.f, 0.f, 0.f, 0.f, 0.f, 0.f, 0.f, 0.f};

  // Mask-free main rounds: all Toeplitz indices provably >= 0.
  const int nFull = t0 >> 4;
  for (int it = 0; it < nFull; ++it) {
    const int sk = (it << 4) + (w << 2) + kb;
    __builtin_prefetch(&uprow[sk + 256], 0, 1); // stream A operand ahead
    v2f a, b;
    a.x = uprow[sk];
    a.y = uprow[sk + 1];
    b.x = ks[t - 1 - sk];
    b.y = ks[t - 2 - sk];
    acc = __builtin_amdgcn_wmma_f32_16x16x4_f32(
        /*neg_a=*/false, a, /*neg_b=*/false, b,
        /*c_mod=*/(short)0, acc, /*reuse_a=*/false, /*reuse_b=*/false);
  }

  // Single masked boundary round: s in [t0, t0+16), causal mask zeroes s >= t.
  {
    const int sk = (nFull << 4) + (w << 2) + kb;  // sk+1 <= t0+15 <= 4095
    v2f a, b;
    a.x = uprow[sk];
    a.y = uprow[sk + 1];
    int k0 = t - 1 - sk;
    int k1 = k0 - 1;
    float b0 = ks[k0 >= 0 ? k0 : 0];
    float b1 = ks[k1 >= 0 ? k1 : 0];
    b.x = (k0 >= 0) ? b0 : 0.0f;
    b.y = (k1 >= 0) ? b1 : 0.0f;
    acc = __builtin_amdgcn_wmma_f32_16x16x4_f32(
        false, a, false, b, (short)0, acc, false, false);
  }

#pragma unroll
  for (int r = 0; r < 8; ++r) part[w][r][lane] = acc[r];
  __syncthreads();

  if (w == 0) {
    const float Dv = Dptr[0];
#pragma unroll
    for (int r = 0; r < 8; ++r) {
      float v = part[0][r][lane] + part[1][r][lane]
              + part[2][r][lane] + part[3][r][lane];
      const int row = r + 8 * (lane >> 4); // C/D layout: VGPR r -> M = r + 8*(lane/16)
      const int idx = row * SEQL + t;
      yp[idx] = v + Dv * up[idx];
    }
  }
}

// ---------------------------------------------------------------------------
// Kernel 5: out[row, :] = yprime[row] * W_out.  134 MB write-once stream ->
// non-temporal b128 stores.
// ---------------------------------------------------------------------------
__global__ __launch_bounds__(256) void s4_expand_kernel(
    const float* __restrict__ yp, const float* __restrict__ Wout,
    float* __restrict__ out)
{
  const long gid = (long)blockIdx.x * 256 + threadIdx.x;
  const long e   = gid * 4;
  const int  row = (int)(e >> 9);
  const int  c   = (int)(e & 511);
  const float v  = yp[row];
  const v4f  wv  = *(const v4f*)(Wout + c);
  v4f o;
  o.x = v * wv.x; o.y = v * wv.y; o.z = v * wv.z; o.w = v * wv.w;
  __builtin_nontemporal_store(o, (v4f*)(out + e));
}

// ---------------------------------------------------------------------------
extern "C" void kernel_launch(void* const* d_in, const int* in_sizes, int n_in,
                              void* d_out, int out_size, void* d_ws, size_t ws_size,
                              hipStream_t stream) {
  const float* u      = (const float*)d_in[0];  // (16, 4096, 512)
  const float* A      = (const float*)d_in[1];  // (64, 64)
  const float* Bv     = (const float*)d_in[2];  // (64, 1)
  const float* Cv     = (const float*)d_in[3];  // (1, 64)
  const float* Dv     = (const float*)d_in[4];  // (1,)
  const float* log_dt = (const float*)d_in[5];  // (1,)
  const float* Win    = (const float*)d_in[6];  // (1, 512)
  const float* Wout   = (const float*)d_in[7];  // (512, 1)
  float* out = (float*)d_out;

  // Workspace layout (floats)
  float* ws      = (float*)d_ws;
  float* wsUp    = ws;                 // 65536  u_proj[b*L + t]
  float* wsYp    = ws + 65536;         // 65536  yprime
  float* wsK     = ws + 131072;        // 4096   conv kernel k[j]
  float* wsAbar  = ws + 135168;        // 4096
  float* wsP16   = ws + 139264;        // 4096
  float* wsBb    = ws + 143360;        // 64
  float* wsSeeds = ws + 143424;        // 16*64

  s4_setup_kernel<<<1, 256, 0, stream>>>(A, Bv, Cv, log_dt,
                                         wsAbar, wsP16, wsBb, wsSeeds);
  s4_kgen_kernel<<<16, 256, 0, stream>>>(wsAbar, wsP16, wsBb, wsSeeds, wsK);
  s4_uproj_kernel<<<(NBATCH * SEQL * 32) / 256, 256, 0, stream>>>(u, Win, wsUp);
  s4_conv_kernel<<<SEQL / 16, 128, 0, stream>>>(wsUp, wsK, Dv, wsYp);
  s4_expand_kernel<<<(NBATCH * SEQL * DMODEL) / (256 * 4), 256, 0, stream>>>(wsYp, Wout, out);
}